// CTCLoss_84722524880964
// MI455X (gfx1250) — compile-verified
//
#include <hip/hip_runtime.h>
#include <cmath>
#include <stdint.h>

#define CTC_BLANK 1
#define CTC_NEG   (-1e30f)

// 2-term logaddexp via hardware v_exp_f32 / v_log_f32 (used once per block).
__device__ __forceinline__ float lae2(float a, float b) {
    float m = fmaxf(a, b);
    float d = fminf(a, b) - m;          // d <= 0; exp underflows to 0 for NEG
    return m + __logf(1.0f + __expf(d));
}

// Fused 3-term logsumexp: max3 + three parallel v_exp_f32 + one v_log_f32.
// Exactly matches lae(lae(a0,a1),a2) (logsumexp is associative), including
// the all-NEG case (returns NEG + log(3), same as the reference recurrence).
__device__ __forceinline__ float lse3(float a0, float a1, float a2) {
    float m = fmaxf(fmaxf(a0, a1), a2);           // v_max3_f32
    float e = __expf(a0 - m) + __expf(a1 - m) + __expf(a2 - m);
    return m + __logf(e);
}

// One workgroup per batch element. Thread s owns extended-label position s.
// LDS layout (byte offsets into the dynamic LDS block):
//   [0, 2*rowBytes)          : double-buffered emission rows (C floats each)
//   OFF_A .. +4*(S+2)        : alpha buffer A (2 NEG pad slots in front)
//   OFF_B .. +4*(S+2)        : alpha buffer B
//   OFF_X .. +4*S            : extended label sequence (int)
__global__ void __launch_bounds__(1024)
ctc_alpha_kernel(const float* __restrict__ log_prob,
                 const int*   __restrict__ input_lengths,
                 const int*   __restrict__ targets,
                 const int*   __restrict__ target_lengths,
                 float*       __restrict__ ll_out,
                 int B, int C, int L)
{
    extern __shared__ char smem[];
    const int S   = 2 * L + 1;
    const int b   = blockIdx.x;
    const int tid = threadIdx.x;
    const bool w0 = (tid < 32);

    const unsigned rowBytes = (unsigned)C * 4u;
    const unsigned OFF_A = rowBytes * 2u;
    const unsigned OFF_B = OFF_A + 4u * (unsigned)(S + 2);
    const unsigned OFF_X = OFF_B + 4u * (unsigned)(S + 2);

    float* ebuf   = (float*)(smem);
    float* alphaA = (float*)(smem + OFF_A);
    float* alphaB = (float*)(smem + OFF_B);
    int*   extl   = (int*)  (smem + OFF_X);

    // Hardware LDS byte address of the dynamic LDS block: low 32 bits of the
    // generic (flat) address (aperture lives in the high 32 bits on CDNA5).
    const unsigned lds_base = (unsigned)(uintptr_t)(void*)smem;

    const int len = input_lengths[b];   // uniform within the block
    const int tl  = target_lengths[b];

    // Build extended label sequence: blank, y1, blank, y2, ..., blank.
    if (tid < S) {
        int v = CTC_BLANK;
        if (tid & 1) {
            int j = tid >> 1;
            v = (j < tl) ? targets[b * L + j] : CTC_BLANK;
        }
        extl[tid] = v;
    }
    if (tid == 0) {
        alphaA[0] = CTC_NEG; alphaA[1] = CTC_NEG;   // pad slots for s-1, s-2
        alphaB[0] = CTC_NEG; alphaB[1] = CTC_NEG;
    }

    const char*  rowbase   = (const char*)log_prob + (size_t)b * rowBytes;
    const size_t rowStride = (size_t)B * rowBytes;

    // Prologue: async-DMA emission rows t=0 (and t=1 if present) into LDS.
    // One row = 512B = 32 lanes x B128, ASYNCcnt-tracked, off the VGPR path.
    if (w0) {
        unsigned lds_addr = lds_base + (unsigned)tid * 16u;
        uint64_t g        = (uint64_t)(rowbase) + (uint64_t)tid * 16u;
        asm volatile("global_load_async_to_lds_b128 %0, %1, off"
                     :: "v"(lds_addr), "v"(g) : "memory");
        if (len > 1) {
            unsigned lds1 = lds_addr + rowBytes;
            uint64_t g1   = g + rowStride;
            asm volatile("global_load_async_to_lds_b128 %0, %1, off"
                         :: "v"(lds1), "v"(g1) : "memory");
        }
        asm volatile("s_wait_asynccnt 0x0" ::: "memory");
    }
    __syncthreads();    // ebuf rows 0/1 + extl + pads visible

    const int s = tid;
    int es = 0, skip = 0;
    if (s < S) {
        es = extl[s];
        int e2 = (s >= 2) ? extl[s - 2] : CTC_BLANK;
        skip = (es != CTC_BLANK) && (es != e2);
    }

    // Peeled t = 0: alpha0[s] = (s < 2) ? lp[0][ext[s]] : NEG  -> buffer A.
    if (s < S) {
        float e = ebuf[es];
        alphaA[s + 2] = (s < 2) ? e : CTC_NEG;
    }
    __syncthreads();    // alpha0 visible; all reads of ebuf half 0 complete

    for (int t = 1; t < len; ++t) {
        const int cur = t & 1;
        float* aprev = cur ? alphaA : alphaB;
        float* acur  = cur ? alphaB : alphaA;

        // Wave 0 prefetches row t+1 into the other half (last read at t-1,
        // closed by the previous barrier) while all 17 waves compute step t.
        if (w0 && (t + 1) < len) {
            unsigned lds_addr = lds_base + (unsigned)(cur ^ 1) * rowBytes
                                         + (unsigned)tid * 16u;
            uint64_t g = (uint64_t)(rowbase) + (uint64_t)(t + 1) * rowStride
                                             + (uint64_t)tid * 16u;
            asm volatile("global_load_async_to_lds_b128 %0, %1, off"
                         :: "v"(lds_addr), "v"(g) : "memory");
        }

        if (s < S) {
            float e  = ebuf[cur * C + es];
            float a0 = aprev[s + 2];                  // alpha[s]
            float a1 = aprev[s + 1];                  // alpha[s-1] (pad-safe)
            float a2 = skip ? aprev[s] : CTC_NEG;     // alpha[s-2] (pad-safe)
            acur[s + 2] = lse3(a0, a1, a2) + e;
        }

        // Drain the async DMA (wave 0 only); one barrier closes the alpha
        // swap and the emission double-buffer swap.
        if (w0) {
            asm volatile("s_wait_asynccnt 0x0" ::: "memory");
        }
        __syncthreads();
    }

    // Final alpha lives in A if (len-1) is even, else B; the loop's (or
    // prologue's) last barrier already made it visible.
    const float* afinal = ((len - 1) & 1) ? alphaB : alphaA;
    if (tid == 0) {
        float l = lae2(afinal[2 * tl + 2], afinal[2 * tl + 1]);
        ll_out[b] = l;
    }
}

// Fixed-order serial reduction for bitwise-deterministic output.
__global__ void ctc_reduce_kernel(const float* __restrict__ ll,
                                  float* __restrict__ out, int B)
{
    if (blockIdx.x == 0 && threadIdx.x == 0) {
        float sum = 0.0f;
        for (int i = 0; i < B; ++i) sum += ll[i];
        out[0] = -sum;
    }
}

extern "C" void kernel_launch(void* const* d_in, const int* in_sizes, int n_in,
                              void* d_out, int out_size, void* d_ws, size_t ws_size,
                              hipStream_t stream) {
    (void)n_in; (void)out_size; (void)ws_size;
    const float* log_prob       = (const float*)d_in[0];   // [T,B,C] f32
    const int*   input_lengths  = (const int*)  d_in[1];   // [B]
    const int*   targets        = (const int*)  d_in[2];   // [B,L]
    const int*   target_lengths = (const int*)  d_in[3];   // [B]

    const int B = in_sizes[1];
    const int L = in_sizes[2] / B;
    const int C = 128;                        // class count (fixed by reference)
    const int S = 2 * L + 1;

    int threads = ((S + 31) / 32) * 32;       // 544 = 17 wave32s for S=513
    if (threads > 1024) threads = 1024;

    size_t lds = (size_t)C * 4 * 2            // emission double buffer
               + 2 * 4 * (size_t)(S + 2)      // alpha double buffer (+pads)
               + 4 * (size_t)S;               // extended labels

    float* ll = (float*)d_ws;                 // per-batch log-likelihoods
    ctc_alpha_kernel<<<B, threads, lds, stream>>>(
        log_prob, input_lengths, targets, target_lengths, ll, B, C, L);
    ctc_reduce_kernel<<<1, 32, 0, stream>>>(ll, (float*)d_out, B);
}